// TeacherDecoder_2052994367735
// MI455X (gfx1250) — compile-verified
//
#include <hip/hip_runtime.h>
#include <hip/hip_bf16.h>
#include <math.h>

// ---------------------------------------------------------------------------
// Sizes fixed by the reference: B=64, S=1024, H=512, T=64
// ---------------------------------------------------------------------------
#define BB   64
#define SS   1024
#define HH   512
#define TT   64
#define NCAT 2560           // 4*H (gates) + H (dec projection)
#define RS   514            // dec row stride in LDS dwords (TDM pad: 512 + 2)

typedef float v2f __attribute__((ext_vector_type(2)));
typedef float v8f __attribute__((ext_vector_type(8)));
typedef unsigned int u32x4 __attribute__((ext_vector_type(4)));
typedef int i32x4 __attribute__((ext_vector_type(4)));
typedef int i32x8 __attribute__((ext_vector_type(8)));

// ---------------------------------------------------------------------------
// WMMA f32 GEMM:  C[M x N] = A[M x K] * B^T (+ bias[N])
// B is pre-packed into WMMA fragment order: Bp[(K/2) x N] of float2 where
// Bp[p][n] = { Bt[2p][n], Bt[2p+1][n] }  ->  each lane's B fragment is ONE
// coalesced global_load_b64.
// One wave per block computes a 32(M) x 128(N) tile: 2 M-subtiles x 8
// N-subtiles -> 16 v_wmma_f32_16x16x4_f32 per k-step vs 10 b64 loads.
//
// Fragment layout (CDNA5 ISA 7.12.2):
//   A 16x4 : lanes 0-15 -> M, v[0..1] = K {0,1}; lanes 16-31 -> K {2,3}
//   B 4x16 : lanes 0-15 -> N, v[0..1] = K {0,1}; lanes 16-31 -> K {2,3}
//   C 16x16: lane (l&15) -> N, vgpr r -> M = r + 8*(l>>4)
// ---------------------------------------------------------------------------
__global__ __launch_bounds__(32)
void gemm_wmma_f32(const float* __restrict__ A, const v2f* __restrict__ Bp,
                   float* __restrict__ C, const float* __restrict__ bias,
                   int K, int lda, int ldb /*= N*/, int ldc)
{
    const int lane  = threadIdx.x & 31;
    const int m0    = blockIdx.y << 5;     // 32-row tile
    const int n0    = blockIdx.x << 7;     // 128-col tile
    const int nlane = lane & 15;
    const int khalf = (lane >> 4) << 1;    // 0 or 2

    v8f acc[2][8] = {};

    const float* Arow0 = A + (size_t)(m0 + nlane) * lda + khalf;
    const float* Arow1 = Arow0 + (size_t)16 * lda;
    // pair-row for this lane = k/2 + (lane>>4)
    const v2f* Bpl = Bp + (size_t)(lane >> 4) * ldb + n0 + nlane;

    for (int k = 0; k < K; k += 4) {
        if (k + 32 < K) {
            __builtin_prefetch(Arow0 + k + 32, 0, 0);               // global_prefetch_b8
            __builtin_prefetch(Arow1 + k + 32, 0, 0);
        }
        const v2f a0 = *(const v2f*)(Arow0 + k);
        const v2f a1 = *(const v2f*)(Arow1 + k);
        const v2f* brow = Bpl + (size_t)(k >> 1) * ldb;
        #pragma unroll
        for (int j = 0; j < 8; ++j) {
            const v2f b = brow[j << 4];                             // global_load_b64
            acc[0][j] = __builtin_amdgcn_wmma_f32_16x16x4_f32(
                            false, a0, false, b, (short)0, acc[0][j], false, false);
            acc[1][j] = __builtin_amdgcn_wmma_f32_16x16x4_f32(
                            false, a1, false, b, (short)0, acc[1][j], false, false);
        }
    }

    #pragma unroll
    for (int mt = 0; mt < 2; ++mt) {
        const int mbase = m0 + (mt << 4) + ((lane >> 4) << 3);
        #pragma unroll
        for (int j = 0; j < 8; ++j) {
            const int n  = n0 + (j << 4) + nlane;
            const float bv = bias ? bias[n] : 0.0f;
            #pragma unroll
            for (int r = 0; r < 8; ++r)
                C[(size_t)(mbase + r) * ldc + n] = acc[mt][j][r] + bv;
        }
    }
}

// ---------------------------------------------------------------------------
// Weight prep: transpose + pack directly into WMMA B-fragment order.
// ---------------------------------------------------------------------------
__global__ void prep_pack_enc(const float* __restrict__ W1, v2f* __restrict__ Bp)
{
    int idx = blockIdx.x * blockDim.x + threadIdx.x;   // (HH/2)*HH threads
    int p = idx >> 9, n = idx & (HH - 1);
    v2f v;
    v.x = W1[(size_t)n * (2 * HH) + 2 * p];            // W1_enc[n][2p]
    v.y = W1[(size_t)n * (2 * HH) + 2 * p + 1];
    Bp[idx] = v;
}

__global__ void prep_pack_cat(const float* __restrict__ W_hh,
                              const float* __restrict__ W1,
                              v2f* __restrict__ Bp)
{
    int idx = blockIdx.x * blockDim.x + threadIdx.x;   // (HH/2)*NCAT threads
    int p = idx / NCAT, n = idx % NCAT;
    v2f v;
    if (n < 4 * HH) {
        v.x = W_hh[(size_t)n * HH + 2 * p];
        v.y = W_hh[(size_t)n * HH + 2 * p + 1];
    } else {
        const float* r = W1 + (size_t)(n - 4 * HH) * (2 * HH) + HH;
        v.x = r[2 * p];
        v.y = r[2 * p + 1];
    }
    Bp[idx] = v;
}

__global__ void prep_bias(const float* __restrict__ b_ih,
                          const float* __restrict__ b_hh,
                          const float* __restrict__ b1,
                          float* __restrict__ bcat)
{
    int n = blockIdx.x * blockDim.x + threadIdx.x;     // NCAT threads
    bcat[n] = (n < 4 * HH) ? (b_ih[n] + b_hh[n]) : b1[n - 4 * HH];
}

// ---------------------------------------------------------------------------
// LSTM cell pointwise update + copy of previous step's dec projection.
// ---------------------------------------------------------------------------
__global__ __launch_bounds__(256)
void lstm_cell(const float* __restrict__ pre, float* __restrict__ h,
               float* __restrict__ c, float* __restrict__ dec_dst,
               int do_update)
{
    int tid = blockIdx.x * blockDim.x + threadIdx.x;   // B*H threads
    int b = tid >> 9, hh = tid & (HH - 1);
    const float* g = pre + (size_t)b * NCAT;
    if (dec_dst) dec_dst[tid] = g[4 * HH + hh];
    if (do_update) {
        float gi = g[hh];
        float gf = g[HH + hh];
        float gg = g[2 * HH + hh];
        float go = g[3 * HH + hh];
        float si = 1.0f / (1.0f + __expf(-gi));
        float sf = 1.0f / (1.0f + __expf(-gf));
        float so = 1.0f / (1.0f + __expf(-go));
        float cn = sf * c[tid] + si * tanhf(gg);
        c[tid] = cn;
        h[tid] = so * tanhf(cn);
    }
}

// ---------------------------------------------------------------------------
// Score pass: out[b][t][s] = sum_h relu(enc_proj[b][s][h] + dec[t][b][h])*W2[h]
//             + b2, masked to -inf.
//
// dec[b] (64 rows x 2KB, row stride 128KB) is pulled into LDS by the Tensor
// Data Mover: 2D D# (tile 512x64, dim0_stride=B*H) with LDS padding of
// 1 dword per 256 dwords -> row stride 514 words; banks = (2t + h) mod 64,
// conflict-free for 32 consecutive t lanes. Issued by wave 0 (TDM ignores
// EXEC), completion via s_wait_tensorcnt then the workgroup barrier.
// ---------------------------------------------------------------------------
__global__ __launch_bounds__(256)
void score_kernel(const float* __restrict__ enc_proj, const float* __restrict__ dec,
                  const float* __restrict__ W2, const float* __restrict__ b2,
                  const int* __restrict__ mask, float* __restrict__ out)
{
    extern __shared__ float smem[];
    float* decS = smem;                       // TT * RS (padded by TDM)
    float* w2s  = decS + TT * RS;             // HH
    float* eS   = w2s + HH;                   // 4 * HH
    const int b      = blockIdx.x;
    const int schunk = blockIdx.y;            // 8 chunks of 128 s
    const int tid    = threadIdx.x;

    if (tid < 32) {
        // --- Tensor DMA: dec[t][b][:] for t=0..63  ->  decS (padded rows) ---
        unsigned long long ga =
            (unsigned long long)(uintptr_t)(dec + (size_t)b * HH);
        unsigned int lds = (unsigned int)(uintptr_t)decS;   // low 32 = LDS offset
        u32x4 g0;
        g0.x = 1u;                                          // count=1 descriptor
        g0.y = lds;                                         // lds_addr
        g0.z = (unsigned int)(ga & 0xffffffffu);            // global_addr[31:0]
        g0.w = (unsigned int)((ga >> 32) & 0x01ffffffu)     // global_addr[56:32]
               | 0x80000000u;                               // type = 2 ("image")
        i32x8 g1;
        g1[0] = 0x01D20000;        // data_size=4B | pad_en | interval=256dw | amt=1dw
        g1[1] = (int)(512u << 16); // tensor_dim0   = 512
        g1[2] = (int)(64u  << 16); // tensor_dim1   = 64
        g1[3] = (int)(512u << 16); // tile_dim0     = 512
        g1[4] = 64;                // tile_dim1     = 64
        g1[5] = BB * HH;           // tensor_dim0_stride = 32768 elements
        g1[6] = 0;
        g1[7] = 0;
        i32x4 z4 = {0, 0, 0, 0};
#if __clang_major__ >= 23
        i32x8 z8 = {0, 0, 0, 0, 0, 0, 0, 0};
        __builtin_amdgcn_tensor_load_to_lds(g0, g1, z4, z4, z8, 0);
#else
        __builtin_amdgcn_tensor_load_to_lds(g0, g1, z4, z4, 0);
#endif
        __builtin_amdgcn_s_wait_tensorcnt(0);
    }
    for (int i = tid; i < HH; i += 256) w2s[i] = W2[i];
    const float bias2 = b2[0];
    __syncthreads();

    const int t  = tid & 63;
    const int sl = tid >> 6;

    for (int sp = 0; sp < 128; sp += 4) {
        const int s_base = schunk * 128 + sp;
        __syncthreads();
        for (int i = tid; i < 4 * HH; i += 256) {
            int r = i >> 9, hh = i & (HH - 1);
            eS[r * HH + hh] = enc_proj[((size_t)b * SS + s_base + r) * HH + hh];
        }
        __syncthreads();
        const int s = s_base + sl;
        const float* er = eS + sl * HH;
        const float* dr = decS + t * RS;
        float acc = 0.0f;
        #pragma unroll 8
        for (int hh = 0; hh < 256; ++hh)
            acc += fmaxf(er[hh] + dr[hh], 0.0f) * w2s[hh];
        #pragma unroll 8
        for (int hh = 256; hh < 512; ++hh)                 // skip the TDM pad dword
            acc += fmaxf(er[hh] + dr[hh + 1], 0.0f) * w2s[hh];
        float sc = acc + bias2;
        sc = (mask[b * SS + s] == 1) ? sc : -INFINITY;
        out[((size_t)b * TT + t) * SS + s] = sc;
    }
}

// ---------------------------------------------------------------------------
// In-place row softmax over S (one block per (b,t) row).
// ---------------------------------------------------------------------------
__global__ __launch_bounds__(256)
void softmax_kernel(float* __restrict__ out)
{
    __shared__ float red[256];
    float* p = out + (size_t)blockIdx.x * SS;
    const int tid = threadIdx.x;

    float m = -INFINITY;
    for (int i = tid; i < SS; i += 256) m = fmaxf(m, p[i]);
    red[tid] = m; __syncthreads();
    for (int off = 128; off > 0; off >>= 1) {
        if (tid < off) red[tid] = fmaxf(red[tid], red[tid + off]);
        __syncthreads();
    }
    m = red[0]; __syncthreads();

    float vals[4]; float sum = 0.0f;
    #pragma unroll
    for (int j = 0; j < 4; ++j) {
        float e = __expf(p[tid + j * 256] - m);
        vals[j] = e; sum += e;
    }
    red[tid] = sum; __syncthreads();
    for (int off = 128; off > 0; off >>= 1) {
        if (tid < off) red[tid] += red[tid + off];
        __syncthreads();
    }
    const float inv = 1.0f / red[0];
    #pragma unroll
    for (int j = 0; j < 4; ++j) p[tid + j * 256] = vals[j] * inv;
}

// ---------------------------------------------------------------------------
// Host orchestration
// ---------------------------------------------------------------------------
extern "C" void kernel_launch(void* const* d_in, const int* in_sizes, int n_in,
                              void* d_out, int out_size, void* d_ws, size_t ws_size,
                              hipStream_t stream)
{
    (void)in_sizes; (void)n_in; (void)out_size; (void)ws_size;

    const float* enc  = (const float*)d_in[0];
    const float* h0   = (const float*)d_in[1];
    const float* c0   = (const float*)d_in[2];
    const int*   msk  = (const int*)d_in[3];
    // d_in[4] = output_length (compile-time TT=64)
    const float* W_hh = (const float*)d_in[6];
    const float* b_ih = (const float*)d_in[7];
    const float* b_hh = (const float*)d_in[8];
    const float* W1   = (const float*)d_in[9];
    const float* b1   = (const float*)d_in[10];
    const float* W2   = (const float*)d_in[11];
    const float* b2   = (const float*)d_in[12];
    float* out = (float*)d_out;

    char* ws = (char*)d_ws;
    float* enc_proj = (float*)ws; ws += (size_t)BB * SS * HH * sizeof(float);     // 134 MB
    float* dec      = (float*)ws; ws += (size_t)TT * BB * HH * sizeof(float);     //   8 MB
    float* pre      = (float*)ws; ws += (size_t)BB * NCAT * sizeof(float);
    float* hbuf     = (float*)ws; ws += (size_t)BB * HH * sizeof(float);
    float* cbuf     = (float*)ws; ws += (size_t)BB * HH * sizeof(float);
    v2f*   Bp_enc   = (v2f*)ws;   ws += (size_t)(HH / 2) * HH * sizeof(v2f);      //   1 MB
    v2f*   Bp_cat   = (v2f*)ws;   ws += (size_t)(HH / 2) * NCAT * sizeof(v2f);    //   5 MB
    float* biascat  = (float*)ws; ws += (size_t)NCAT * sizeof(float);

    // --- weight prep (transpose + WMMA fragment packing) ---
    prep_pack_enc<<<((HH / 2) * HH) / 256, 256, 0, stream>>>(W1, Bp_enc);
    prep_pack_cat<<<((HH / 2) * NCAT) / 256, 256, 0, stream>>>(W_hh, W1, Bp_cat);
    prep_bias<<<NCAT / 256, 256, 0, stream>>>(b_ih, b_hh, b1, biascat);
    hipMemcpyAsync(hbuf, h0, (size_t)BB * HH * sizeof(float),
                   hipMemcpyDeviceToDevice, stream);
    hipMemcpyAsync(cbuf, c0, (size_t)BB * HH * sizeof(float),
                   hipMemcpyDeviceToDevice, stream);

    // --- enc_proj = E @ W1_enc^T  (M=65536, N=512, K=512) ---
    dim3 ge(HH / 128, (BB * SS) / 32);
    gemm_wmma_f32<<<ge, 32, 0, stream>>>(enc, Bp_enc, enc_proj, nullptr,
                                         HH, HH, HH, HH);

    // --- LSTM: per step one fused GEMM h @ [W_hh ; W1_dec]^T ---
    dim3 gs(NCAT / 128, BB / 32);
    gemm_wmma_f32<<<gs, 32, 0, stream>>>(hbuf, Bp_cat, pre, biascat,
                                         HH, HH, NCAT, NCAT);   // gates_0
    for (int t = 0; t < TT; ++t) {
        lstm_cell<<<(BB * HH) / 256, 256, 0, stream>>>(
            pre, hbuf, cbuf,
            t > 0 ? dec + (size_t)(t - 1) * BB * HH : nullptr, 1);
        gemm_wmma_f32<<<gs, 32, 0, stream>>>(hbuf, Bp_cat, pre, biascat,
                                             HH, HH, NCAT, NCAT);
    }
    lstm_cell<<<(BB * HH) / 256, 256, 0, stream>>>(
        pre, hbuf, cbuf, dec + (size_t)(TT - 1) * BB * HH, 0);

    // --- scores + mask (TDM-staged dec in LDS) ---
    dim3 gsc(BB, SS / 128);
    size_t smem = (size_t)(TT * RS + HH + 4 * HH) * sizeof(float);
    score_kernel<<<gsc, 256, smem, stream>>>(enc_proj, dec, W2, b2, msk, out);

    // --- softmax over S ---
    softmax_kernel<<<BB * TT, 256, 0, stream>>>(out);
}